// dsfa_former_24386824306982
// MI455X (gfx1250) — compile-verified
//
#include <hip/hip_runtime.h>
#include <math.h>

// ---------------------------------------------------------------------------
// dsfa_former hybrid attention on MI455X (gfx1250), fp32 via V_WMMA_F32_16X16X4_F32
//
// B=4, N=2048, H=8, d=32, kproj=256.  ~30 GFLOP fp32; working set ~30MB fits in
// the 192MB L2, so the optimization target is instruction economy + coalescing:
//  - flash attention: 4 waves/block share one (b,h) and a 64-row query
//    super-tile; each 16x32 K/V tile is staged to LDS ONCE per block as one
//    b128 per thread (coalesced), via the gfx1250 async-to-LDS path (ASYNCcnt)
//    when the builtin is available, then WMMA fragments are read from LDS
//    bank-conflict-free.
//  - linproj: each wave computes a 2x2 tile grid so every loaded fragment
//    feeds 2 WMMAs (2 loads/WMMA instead of 4).
//
// WMMA f32 16x16x4 lane layouts (CDNA5 ISA 7.12.2, wave32):
//   A (16x4, MxK):  lane l holds A[l&15][koff], A[l&15][koff+1], koff = 2*(l>>4)
//   B (4x16, KxN):  lane l holds B[koff][l&15], B[koff+1][l&15]
//   C/D (16x16):    VGPR i of lane l = D[i + 8*(l>>4)][l&15]
// ---------------------------------------------------------------------------

typedef float v2f __attribute__((ext_vector_type(2)));
typedef float v8f __attribute__((ext_vector_type(8)));
typedef int   v4i __attribute__((vector_size(16)));   // matches builtin proto

#define WMMA_F32(a, b, c) \
  __builtin_amdgcn_wmma_f32_16x16x4_f32(false, (a), false, (b), (short)0, (c), false, false)

static constexpr int B_  = 4;
static constexpr int N_  = 2048;
static constexpr int H_  = 8;
static constexpr int D_  = 32;    // head dim
static constexpr int KP_ = 256;   // Linformer projection length

// LDS row pads (floats). KV pad 44: b64 K-frag reads: 44*dl mod 64 never in
// {1,2,3,61,62,63} and 44/4=11 odd => 16 distinct banks per half => 64 distinct
// dwords across the wave. V b32 reads: halves differ by 2 rows => bank delta
// 88=24 mod 64, needs lane delta 24 to collide => impossible. Pad must be a
// multiple of 4 for b128 staging alignment.
static constexpr int KVP = 44;
// P-tile pad 18 (even => aligned b64 A-frag reads; 18*dl mod 64 distinct,
// cross-half collision needs 9*dl == 1 mod 32 => dl = -7, so 2-way conflicts occur).
static constexpr int PP  = 18;

#if defined(__gfx1250__) && __has_builtin(__builtin_amdgcn_global_load_async_to_lds_b128)
#define HAVE_ASYNC_LDS 1
#endif

__device__ __forceinline__ void stage_b128(const float* __restrict__ g, float* l) {
#if defined(HAVE_ASYNC_LDS)
  __builtin_amdgcn_global_load_async_to_lds_b128(
      (__attribute__((address_space(1))) v4i*)g,
      (__attribute__((address_space(3))) v4i*)l, 0, 0);
#else
  *(float4*)l = *(const float4*)g;
#endif
}

__device__ __forceinline__ void stage_wait() {
#if defined(HAVE_ASYNC_LDS)
#if __has_builtin(__builtin_amdgcn_s_wait_asynccnt)
  __builtin_amdgcn_s_wait_asynccnt(0);
#else
  asm volatile("s_wait_asynccnt 0x0" ::: "memory");
#endif
#endif
}

// ---------------------------------------------------------------------------
// Kernel 1: Linformer projections  Kp[bh][kp][d] = sum_n E[n][kp] * K[b][n][h][d]
// (and Vp with F).  One wave per 32(kp) x 32(d) output block = 2x2 WMMA tiles;
// each E/F fragment is reused across both d-tiles, each K/V fragment across
// both kp-tiles -> 16 b32 loads per 8 WMMAs.
// ---------------------------------------------------------------------------
__global__ __launch_bounds__(128) void linproj_kernel(
    const float* __restrict__ K, const float* __restrict__ V,
    const float* __restrict__ E, const float* __restrict__ F,
    float* __restrict__ Kp, float* __restrict__ Vp)
{
  const int lane = threadIdx.x & 31;
  const int wid  = threadIdx.x >> 5;
  const int w    = blockIdx.x * 4 + wid;      // 0..255 wave-tasks
  const int bh   = w >> 3;                    // 8 kp-pair tasks per (b,h)
  const int kpp  = w & 7;
  const int kp0  = kpp * 32;                  // kp tile pair [kp0, kp0+32)
  const int b = bh >> 3, h = bh & 7;

  const int half = lane >> 4;
  const int ln   = lane & 15;
  const int koff = half * 2;

  v8f cK00 = {}, cK01 = {}, cK10 = {}, cK11 = {};
  v8f cV00 = {}, cV01 = {}, cV10 = {}, cV11 = {};

  for (int nb = 0; nb < N_; nb += 4) {
    const int r0 = nb + koff;
    // A = E^T / F^T : A[m][k] = E[nb+k][kp0+m]
    const float* e0 = E + (size_t)r0 * KP_ + kp0 + ln;
    const float* f0 = F + (size_t)r0 * KP_ + kp0 + ln;
    v2f ae0, ae1, af0, af1;
    ae0.x = e0[0];  ae0.y = e0[KP_];
    ae1.x = e0[16]; ae1.y = e0[KP_ + 16];
    af0.x = f0[0];  af0.y = f0[KP_];
    af1.x = f0[16]; af1.y = f0[KP_ + 16];
    // B[k][n] = K[b][nb+k][h][n] ; consecutive n-rows stride H_*D_
    const float* kr = K + ((size_t)(b * N_ + r0) * H_ + h) * D_ + ln;
    const float* vr = V + ((size_t)(b * N_ + r0) * H_ + h) * D_ + ln;
    v2f bk0, bk1, bv0, bv1;
    bk0.x = kr[0];  bk0.y = kr[H_ * D_];
    bk1.x = kr[16]; bk1.y = kr[H_ * D_ + 16];
    bv0.x = vr[0];  bv0.y = vr[H_ * D_];
    bv1.x = vr[16]; bv1.y = vr[H_ * D_ + 16];

    cK00 = WMMA_F32(ae0, bk0, cK00);  cK01 = WMMA_F32(ae0, bk1, cK01);
    cK10 = WMMA_F32(ae1, bk0, cK10);  cK11 = WMMA_F32(ae1, bk1, cK11);
    cV00 = WMMA_F32(af0, bv0, cV00);  cV01 = WMMA_F32(af0, bv1, cV01);
    cV10 = WMMA_F32(af1, bv0, cV10);  cV11 = WMMA_F32(af1, bv1, cV11);
  }

  float* kpOut = Kp + ((size_t)bh * KP_ + kp0) * D_;
  float* vpOut = Vp + ((size_t)bh * KP_ + kp0) * D_;
#pragma unroll
  for (int i = 0; i < 8; ++i) {
    const int m = i + 8 * half;
    kpOut[(size_t)m * D_ + ln]             = cK00[i];
    kpOut[(size_t)m * D_ + 16 + ln]        = cK01[i];
    kpOut[(size_t)(m + 16) * D_ + ln]      = cK10[i];
    kpOut[(size_t)(m + 16) * D_ + 16 + ln] = cK11[i];
    vpOut[(size_t)m * D_ + ln]             = cV00[i];
    vpOut[(size_t)m * D_ + 16 + ln]        = cV01[i];
    vpOut[(size_t)(m + 16) * D_ + ln]      = cV10[i];
    vpOut[(size_t)(m + 16) * D_ + 16 + ln] = cV11[i];
  }
}

// ---------------------------------------------------------------------------
// Kernel 2: flash attention.  4 waves/block = one 64-row query super-tile of
// one (b,h); the block stages each 16x32 K and V tile into LDS once
// (one b128 per thread, coalesced; async-to-LDS when available), all waves
// consume it.  Wave w owns query rows [qs0 + 16w, qs0 + 16w + 16).
// Q pre-scaled by 1/sqrt(d): softmax(scale*QK^T) == softmax((scale*Q)K^T).
// ---------------------------------------------------------------------------
__global__ __launch_bounds__(128) void flash16_kernel(
    const float* __restrict__ Q,
    const float* __restrict__ K,
    const float* __restrict__ V,
    float* __restrict__ Out,
    int kvLen, int kvRowStride, int kvBaseB, int kvBaseH,
    int causal, float scale)
{
  __shared__ float sK[16 * KVP];
  __shared__ float sV[16 * KVP];
  __shared__ float pT[4][16 * PP];

  const int lane = threadIdx.x & 31;
  const int wid  = threadIdx.x >> 5;
  const int bh   = blockIdx.x >> 5;        // 32 super-tiles per (b,h)
  const int st   = blockIdx.x & 31;
  const int b = bh >> 3, h = bh & 7;
  const int qtWave = st * 4 + wid;         // this wave's 16-row q tile index
  const int qr0 = qtWave * 16;

  const int half = lane >> 4;
  const int ln   = lane & 15;
  const int koff = half * 2;

  // Q fragments (A-layout), pre-scaled.
  const float* qrow = Q + ((size_t)(b * N_ + qr0 + ln) * H_ + h) * D_;
  v2f qf[8];
#pragma unroll
  for (int kk = 0; kk < 8; ++kk) {
    v2f q = *(const v2f*)(qrow + kk * 4 + koff);
    qf[kk] = q * scale;
  }

  const float* Kbh = K + (size_t)b * kvBaseB + (size_t)h * kvBaseH;
  const float* Vbh = V + (size_t)b * kvBaseB + (size_t)h * kvBaseH;

  v8f o0 = {}, o1 = {};              // output d-cols [0,16) and [16,32)
  float mrow[8], lrow[8];
#pragma unroll
  for (int i = 0; i < 8; ++i) { mrow[i] = -__builtin_inff(); lrow[i] = 0.0f; }

  const int jTiles = causal ? (st * 4 + 4) : (kvLen >> 4);
  float* myP = pT[wid];

  // staging coordinates: thread t loads K/V[jb + t/8][4*(t%8) .. +3]
  const int srow = threadIdx.x >> 3;
  const int scol = (threadIdx.x & 7) * 4;

  for (int j = 0; j < jTiles; ++j) {
    const int jb = j * 16;

    // ---- cooperative tile staging (coalesced b128 per thread) ----
    const float* gk = Kbh + (size_t)(jb + srow) * kvRowStride + scol;
    const float* gv = Vbh + (size_t)(jb + srow) * kvRowStride + scol;
    stage_b128(gk, &sK[srow * KVP + scol]);
    stage_b128(gv, &sV[srow * KVP + scol]);
    stage_wait();
    __syncthreads();

    const bool active = (!causal) || (j <= qtWave);   // wave-uniform
    if (active) {
      // ---- S = (scale*Q) * K^T : 8 WMMAs over d=32 ----
      v8f s = {};
#pragma unroll
      for (int kk = 0; kk < 8; ++kk) {
        v2f bf = *(const v2f*)&sK[ln * KVP + kk * 4 + koff];  // K[jb+ln][..]
        s = WMMA_F32(qf[kk], bf, s);
      }

      // ---- causal mask (diagonal tile only) ----
      if (causal && j == qtWave) {
        const int kc = jb + ln;
#pragma unroll
        for (int i = 0; i < 8; ++i) {
          const int qr = qr0 + i + 8 * half;
          if (kc > qr) s[i] = -__builtin_inff();
        }
      }

      // ---- online softmax; row m lives across 16 lanes of one half ----
      float corr[8];
#pragma unroll
      for (int i = 0; i < 8; ++i) {
        float rmax = s[i];
        rmax = fmaxf(rmax, __shfl_xor(rmax, 1));
        rmax = fmaxf(rmax, __shfl_xor(rmax, 2));
        rmax = fmaxf(rmax, __shfl_xor(rmax, 4));
        rmax = fmaxf(rmax, __shfl_xor(rmax, 8));
        const float mnew = fmaxf(mrow[i], rmax);
        const float c = __expf(mrow[i] - mnew);   // exp(-inf)=0 on first tile
        const float p = __expf(s[i] - mnew);      // masked -> 0
        s[i] = p;
        float rsum = p;
        rsum += __shfl_xor(rsum, 1);
        rsum += __shfl_xor(rsum, 2);
        rsum += __shfl_xor(rsum, 4);
        rsum += __shfl_xor(rsum, 8);
        lrow[i] = lrow[i] * c + rsum;
        mrow[i] = mnew;
        corr[i] = c;
      }
#pragma unroll
      for (int i = 0; i < 8; ++i) { o0[i] *= corr[i]; o1[i] *= corr[i]; }

      // ---- stage P (D-layout) into wave-private LDS tile ----
#pragma unroll
      for (int i = 0; i < 8; ++i)
        myP[(i + 8 * half) * PP + ln] = s[i];
      asm volatile("" ::: "memory");   // LDS in-order per wave; block reordering

      // ---- O += P * V : 4 K-chunks x 2 d-tiles = 8 WMMAs ----
#pragma unroll
      for (int kk = 0; kk < 4; ++kk) {
        v2f pa = *(const v2f*)&myP[ln * PP + kk * 4 + koff];
        const int r = kk * 4 + koff;
        v2f bv0, bv1;
        bv0.x = sV[r * KVP + ln];
        bv0.y = sV[(r + 1) * KVP + ln];
        bv1.x = sV[r * KVP + 16 + ln];
        bv1.y = sV[(r + 1) * KVP + 16 + ln];
        o0 = WMMA_F32(pa, bv0, o0);
        o1 = WMMA_F32(pa, bv1, o1);
      }
      asm volatile("" ::: "memory");
    }
    __syncthreads();   // everyone done with sK/sV before next staging
  }

  // ---- normalize and store [B,N,H,d] ----
  float* orow = Out + ((size_t)(b * N_ + qr0) * H_ + h) * D_;
#pragma unroll
  for (int i = 0; i < 8; ++i) {
    const int m = i + 8 * half;
    const float inv = 1.0f / lrow[i];
    orow[(size_t)m * (H_ * D_) + ln]      = o0[i] * inv;
    orow[(size_t)m * (H_ * D_) + 16 + ln] = o1[i] * inv;
  }
}

// ---------------------------------------------------------------------------
extern "C" void kernel_launch(void* const* d_in, const int* in_sizes, int n_in,
                              void* d_out, int out_size, void* d_ws, size_t ws_size,
                              hipStream_t stream)
{
  (void)in_sizes; (void)n_in; (void)out_size; (void)ws_size;
  const float* Q = (const float*)d_in[0];
  const float* K = (const float*)d_in[1];
  const float* V = (const float*)d_in[2];
  const float* E = (const float*)d_in[3];
  const float* F = (const float*)d_in[4];

  float* outFull = (float*)d_out;                          // [B,N,H,d]
  float* outLin  = outFull + (size_t)B_ * N_ * H_ * D_;    // [B,N,H,d]

  float* Kp = (float*)d_ws;                                // [32, 256, 32] = 1 MB
  float* Vp = Kp + (size_t)(B_ * H_) * KP_ * D_;           // [32, 256, 32] = 1 MB

  const float scale = 0.17677669529663687f;                // 1/sqrt(32)

  // (1) Linformer projections: 256 wave-tasks, 4 waves/block
  linproj_kernel<<<64, 128, 0, stream>>>(K, V, E, F, Kp, Vp);

  // (2) Full causal attention: 32 bh * 32 super-tiles = 1024 blocks
  flash16_kernel<<<1024, 128, 0, stream>>>(
      Q, K, V, outFull,
      /*kvLen*/ N_, /*rowStride*/ H_ * D_,
      /*baseB*/ N_ * H_ * D_, /*baseH*/ D_,
      /*causal*/ 1, scale);

  // (3) Linformer attention over projected K/V (kvLen=256, no mask)
  flash16_kernel<<<1024, 128, 0, stream>>>(
      Q, Kp, Vp, outLin,
      /*kvLen*/ KP_, /*rowStride*/ D_,
      /*baseB*/ H_ * KP_ * D_, /*baseH*/ KP_ * D_,
      /*causal*/ 0, scale);
}